// RoI2Det_128849019031
// MI455X (gfx1250) — compile-verified
//
#include <hip/hip_runtime.h>
#include <hip/hip_bf16.h>
#include <math.h>

// ---------------- problem constants (match reference) ----------------
#define BATCH   4
#define NPROP   2000
#define NCLS    80
#define NCLS1   81
#define KPRE    2000
#define KSORT   2048          // padded selection size (2048 = 64*32)
#define NT32    64            // 32-wide tiles over KSORT
#define MROWS   2048          // mask rows (padded)
#define MAXDET  100
#define SCORE_THR 0.05f
#define IOU_THR   0.5f
#define MAX_RATIO 4.135166556742356f   // |log(16/1000)|

typedef __attribute__((ext_vector_type(2))) float v2f;
typedef __attribute__((ext_vector_type(8))) float v8f;

#if __has_builtin(__builtin_amdgcn_tensor_load_to_lds) && __has_builtin(__builtin_amdgcn_s_wait_tensorcnt)
#define USE_TDM 1
typedef __attribute__((ext_vector_type(4))) unsigned int u32x4;
typedef __attribute__((ext_vector_type(4))) int i32x4;
typedef __attribute__((ext_vector_type(8))) int i32x8;
#endif

// ---------------- kernel 1: per-row softmax stats (max, sumexp) ------
__global__ void rowstat_kernel(const float* __restrict__ cls,
                               float* __restrict__ rowMax,
                               float* __restrict__ rowSum) {
    int t = blockIdx.x * blockDim.x + threadIdx.x;
    if (t >= BATCH * NPROP) return;
    const float* p = cls + (size_t)t * NCLS1;
    __builtin_prefetch(p, 0, 1);            // gfx1250 global_prefetch_b8
    float m = p[0];
    #pragma unroll 9
    for (int i = 1; i < NCLS1; ++i) m = fmaxf(m, p[i]);
    float s = 0.f;
    #pragma unroll 9
    for (int i = 0; i < NCLS1; ++i) s += expf(p[i] - m);
    rowMax[t] = m;
    rowSum[t] = s;
}

// ---------------- kernel 2: softmax score + bbox decode --------------
__global__ void score_decode_kernel(const float* __restrict__ cls,
                                    const float* __restrict__ reg,
                                    const float* __restrict__ props,
                                    const int*   __restrict__ hw,
                                    const float* __restrict__ rowMax,
                                    const float* __restrict__ rowSum,
                                    float* __restrict__ cand,
                                    float* __restrict__ fbox) {
    int t = blockIdx.x * blockDim.x + threadIdx.x;
    if (t >= BATCH * NPROP * NCLS) return;
    int c  = t % NCLS;
    int bn = t / NCLS;
    int b  = bn / NPROP;

    float sc = expf(cls[(size_t)bn * NCLS1 + c] - rowMax[bn]) / rowSum[bn];
    cand[t] = (sc > SCORE_THR) ? sc : -1.0f;

    const float* pr = props + (size_t)bn * 4;
    const float* dl = reg   + (size_t)bn * NCLS * 4 + (size_t)c * 4;
    float H = (float)hw[b * 2 + 0];
    float W = (float)hw[b * 2 + 1];
    float dx = dl[0] * 0.1f;
    float dy = dl[1] * 0.1f;
    float dw = fminf(fmaxf(dl[2] * 0.2f, -MAX_RATIO), MAX_RATIO);
    float dh = fminf(fmaxf(dl[3] * 0.2f, -MAX_RATIO), MAX_RATIO);
    float px = (pr[0] + pr[2]) * 0.5f;
    float py = (pr[1] + pr[3]) * 0.5f;
    float pw = pr[2] - pr[0];
    float ph = pr[3] - pr[1];
    float gx = px + pw * dx;
    float gy = py + ph * dy;
    float gw = pw * expf(dw);
    float gh = ph * expf(dh);
    float* o = fbox + (size_t)t * 4;
    o[0] = fminf(fmaxf(gx - gw * 0.5f, 0.f), W);
    o[1] = fminf(fmaxf(gy - gh * 0.5f, 0.f), H);
    o[2] = fminf(fmaxf(gx + gw * 0.5f, 0.f), W);
    o[3] = fminf(fmaxf(gy + gh * 0.5f, 0.f), H);
}

// ---------------- kernel 3: per-image top-K select + bitonic sort ----
__global__ __launch_bounds__(1024)
void select_kernel(const float* __restrict__ cand,
                   const float* __restrict__ fbox,
                   float* __restrict__ selScore,
                   float* __restrict__ selBox,
                   int*   __restrict__ selLabel) {
    const int b = blockIdx.x;
    const int tid = threadIdx.x;
    const int TOT = NPROP * NCLS;
    __shared__ unsigned hist[4096];
    __shared__ float sScore[KSORT];
    __shared__ int   sIdx[KSORT];
    __shared__ int sT, sT2, sTake, sAbove, sAcceptAll;
    __shared__ unsigned cntHigh, cntBnd;

    const float* cs = cand + (size_t)b * TOT;

    for (int i = tid; i < 4096; i += 1024) hist[i] = 0;
    if (tid == 0) { cntHigh = 0; cntBnd = 0; sT = 0; sT2 = 0; sTake = 0; sAbove = 0; sAcceptAll = 0; }
    __syncthreads();

    // pass 1: coarse histogram on float bits [30:19] (scores in (0,1])
    for (int f = tid; f < TOT; f += 1024) {
        float s = cs[f];
        if (s > 0.f) atomicAdd(&hist[(__float_as_uint(s) >> 19) & 0xFFFu], 1u);
    }
    __syncthreads();
    if (tid == 0) {
        // scores <= 1.0f  =>  key <= 0x7F0 = 2032; start just above
        unsigned cum = 0; int t = -1; unsigned above = 0;
        for (int bin = 2047; bin >= 0; --bin) {
            unsigned c2 = cum + hist[bin];
            if (c2 >= KPRE) { t = bin; above = cum; break; }
            cum = c2;
        }
        sT = t; sAbove = (int)above; sAcceptAll = (t < 0) ? 1 : 0;
    }
    __syncthreads();
    const int T = sT;
    const int acceptAll = sAcceptAll;

    if (!acceptAll) {
        // pass 2: 256-bin refine on float bits [18:11] within boundary bin
        for (int i = tid; i < 256; i += 1024) hist[i] = 0;
        __syncthreads();
        for (int f = tid; f < TOT; f += 1024) {
            float s = cs[f];
            if (s > 0.f) {
                unsigned bits = __float_as_uint(s);
                if (((bits >> 19) & 0xFFFu) == (unsigned)T)
                    atomicAdd(&hist[(bits >> 11) & 0xFFu], 1u);
            }
        }
        __syncthreads();
        if (tid == 0) {
            unsigned cum = (unsigned)sAbove;
            for (int bin = 255; bin >= 0; --bin) {
                unsigned c2 = cum + hist[bin];
                if (c2 >= KPRE) { sT2 = bin; sAbove = (int)cum; sTake = (int)(KPRE - cum); break; }
                cum = c2;
            }
        }
        __syncthreads();
    }
    const int T2 = sT2, TAKE = sTake, ABOVE = sAbove;

    for (int i = tid; i < KSORT; i += 1024) { sScore[i] = -1e30f; sIdx[i] = 0; }
    __syncthreads();

    // compaction (slot order irrelevant: sorted below)
    for (int f = tid; f < TOT; f += 1024) {
        float s = cs[f];
        if (s <= 0.f) continue;
        unsigned bits = __float_as_uint(s);
        int k  = (int)((bits >> 19) & 0xFFFu);
        int k2 = (int)((bits >> 11) & 0xFFu);
        int slot = -1;
        if (acceptAll || k > T || (k == T && k2 > T2)) {
            slot = (int)atomicAdd(&cntHigh, 1u);
        } else if (k == T && k2 == T2) {
            unsigned p = atomicAdd(&cntBnd, 1u);
            if ((int)p < TAKE) slot = ABOVE + (int)p;
        }
        if (slot >= 0 && slot < KSORT) { sScore[slot] = s; sIdx[slot] = f; }
    }
    __syncthreads();

    // bitonic sort, descending by score
    for (unsigned k = 2; k <= KSORT; k <<= 1) {
        for (unsigned j = k >> 1; j > 0; j >>= 1) {
            for (unsigned i = tid; i < KSORT; i += 1024) {
                unsigned ixj = i ^ j;
                if (ixj > i) {
                    bool up = ((i & k) == 0);
                    float a = sScore[i], c2 = sScore[ixj];
                    bool swp = up ? (a < c2) : (a > c2);
                    if (swp) {
                        sScore[i] = c2; sScore[ixj] = a;
                        int tmp = sIdx[i]; sIdx[i] = sIdx[ixj]; sIdx[ixj] = tmp;
                    }
                }
            }
            __syncthreads();
        }
    }

    // emit selected candidates (score, box, label); padding rows are inert
    for (int j = tid; j < KSORT; j += 1024) {
        float s = sScore[j];
        int f = sIdx[j];
        float* ob = selBox + ((size_t)b * KSORT + j) * 4;
        if (s > SCORE_THR) {
            selScore[b * KSORT + j] = s;
            const float* ib = fbox + ((size_t)b * TOT + f) * 4;
            ob[0] = ib[0]; ob[1] = ib[1]; ob[2] = ib[2]; ob[3] = ib[3];
            selLabel[b * KSORT + j] = f % NCLS;
        } else {
            selScore[b * KSORT + j] = -1.f;
            ob[0] = ob[1] = ob[2] = ob[3] = 0.f;
            selLabel[b * KSORT + j] = -1;
        }
    }
}

// -------- kernel 4: 32x32 IoU region per wave32, 4 WMMAs -------------
// Per 16x16 sub-tile: D = A(16x4) x B(4x16) + C, with A col0 = area_i,
// col1 = 1; B row0 = 1, row1 = area_j; C = 1e-6  ->
// D[m][n] = area_i[m] + area_j[n] + eps.  Then
// iou > t  <=>  inter > t * (D - inter)   (no division needed).
__global__ __launch_bounds__(32)
void iou_mask_kernel(const float* __restrict__ selBox,
                     const int*   __restrict__ selLabel,
                     const int*   __restrict__ hw,
                     unsigned*    __restrict__ mask) {
    const int ti = blockIdx.x, tj = blockIdx.y, b = blockIdx.z;
    if (tj < ti) return;                       // uniform over the wave
    const int lane = threadIdx.x;

    __shared__ float rB[32 * 4], cB[32 * 4], rA[32], cA[32];
#if defined(USE_TDM)
    __shared__ float rawR[128], rawC[128];
#endif

    float H = (float)hw[b * 2 + 0], W = (float)hw[b * 2 + 1];
    float off_scale = fmaxf(H, W) + 1.0f;      // class-offset trick

    const size_t rowBase = (size_t)b * KSORT + (size_t)ti * 32;
    const size_t colBase = (size_t)b * KSORT + (size_t)tj * 32;

#if defined(USE_TDM)
    {
        // TDM: stage the two contiguous 512B box chunks into LDS.
        // D# group0: [1:0]=count=1, [63:32]=lds_addr, [120:64]=global_addr,
        //            [127:126]=type=2.  group1: data_size=4B, 1-D tile of
        //            128 dwords (tensor_dim0=tile_dim0=128, dim1=1, stride=128).
        // amdgpu-toolchain (clang-23) 6-arg form: (g0, g1, g2, g3, g4, cpol).
        unsigned ldsR = (unsigned)(uintptr_t)(&rawR[0]);
        unsigned ldsC = (unsigned)(uintptr_t)(&rawC[0]);
        unsigned long long gR = (unsigned long long)(uintptr_t)(selBox + rowBase * 4);
        unsigned long long gC = (unsigned long long)(uintptr_t)(selBox + colBase * 4);
        i32x8 g1;
        g1[0] = 0x00020000;          // data_size=2 (4 bytes), wg_mask=0
        g1[1] = (int)(128u << 16);   // tensor_dim0 lo16 in [63:48]
        g1[2] = (int)(1u << 16);     // tensor_dim0 hi16=0, tensor_dim1 lo16=1
        g1[3] = (int)(128u << 16);   // tensor_dim1 hi16=0, tile_dim0=128
        g1[4] = 1;                   // tile_dim1=1, tile_dim2=0
        g1[5] = 128;                 // tensor_dim0_stride lo32
        g1[6] = 0;
        g1[7] = 0;
        i32x4 gz; gz[0] = 0; gz[1] = 0; gz[2] = 0; gz[3] = 0;
        i32x8 gz8;
        gz8[0] = 0; gz8[1] = 0; gz8[2] = 0; gz8[3] = 0;
        gz8[4] = 0; gz8[5] = 0; gz8[6] = 0; gz8[7] = 0;
        u32x4 g0r; g0r[0] = 1u; g0r[1] = ldsR;
        g0r[2] = (unsigned)gR; g0r[3] = (unsigned)(gR >> 32) | (2u << 30);
        u32x4 g0c; g0c[0] = 1u; g0c[1] = ldsC;
        g0c[2] = (unsigned)gC; g0c[3] = (unsigned)(gC >> 32) | (2u << 30);
        __builtin_amdgcn_tensor_load_to_lds(g0r, g1, gz, gz, gz8, 0);
        __builtin_amdgcn_tensor_load_to_lds(g0c, g1, gz, gz, gz8, 0);
        __builtin_amdgcn_s_wait_tensorcnt(0);
        __syncthreads();
    }
    {
        float oR = (float)selLabel[rowBase + lane] * off_scale;
        float x1 = rawR[lane * 4 + 0] + oR, y1 = rawR[lane * 4 + 1] + oR;
        float x2 = rawR[lane * 4 + 2] + oR, y2 = rawR[lane * 4 + 3] + oR;
        rB[lane * 4 + 0] = x1; rB[lane * 4 + 1] = y1;
        rB[lane * 4 + 2] = x2; rB[lane * 4 + 3] = y2;
        rA[lane] = (x2 - x1) * (y2 - y1);
        float oC = (float)selLabel[colBase + lane] * off_scale;
        float u1 = rawC[lane * 4 + 0] + oC, v1 = rawC[lane * 4 + 1] + oC;
        float u2 = rawC[lane * 4 + 2] + oC, v2 = rawC[lane * 4 + 3] + oC;
        cB[lane * 4 + 0] = u1; cB[lane * 4 + 1] = v1;
        cB[lane * 4 + 2] = u2; cB[lane * 4 + 3] = v2;
        cA[lane] = (u2 - u1) * (v2 - v1);
    }
#else
    {
        const float* pr = selBox + rowBase * 4 + lane * 4;
        float oR = (float)selLabel[rowBase + lane] * off_scale;
        float x1 = pr[0] + oR, y1 = pr[1] + oR, x2 = pr[2] + oR, y2 = pr[3] + oR;
        rB[lane * 4 + 0] = x1; rB[lane * 4 + 1] = y1;
        rB[lane * 4 + 2] = x2; rB[lane * 4 + 3] = y2;
        rA[lane] = (x2 - x1) * (y2 - y1);
        const float* pc = selBox + colBase * 4 + lane * 4;
        float oC = (float)selLabel[colBase + lane] * off_scale;
        float u1 = pc[0] + oC, v1 = pc[1] + oC, u2 = pc[2] + oC, v2 = pc[3] + oC;
        cB[lane * 4 + 0] = u1; cB[lane * 4 + 1] = v1;
        cB[lane * 4 + 2] = u2; cB[lane * 4 + 3] = v2;
        cA[lane] = (u2 - u1) * (v2 - v1);
    }
#endif
    __syncthreads();

    const int nn = lane & 15;
    const int half = lane >> 4;

    #pragma unroll
    for (int si = 0; si < 2; ++si) {
        #pragma unroll
        for (int sj = 0; sj < 2; ++sj) {
            // A 16x4 layout: lanes 0-15 hold (K0,K1); lanes 16-31 = (K2,K3)=0
            v2f a, bf;
            a.x  = (lane < 16) ? rA[si * 16 + lane] : 0.f;   // K0 = area_i
            a.y  = (lane < 16) ? 1.f                : 0.f;   // K1 = 1
            bf.x = (lane < 16) ? 1.f                : 0.f;   // K0 row = ones
            bf.y = (lane < 16) ? cA[sj * 16 + lane] : 0.f;   // K1 row = area_j
            v8f acc = {1e-6f, 1e-6f, 1e-6f, 1e-6f, 1e-6f, 1e-6f, 1e-6f, 1e-6f};
            acc = __builtin_amdgcn_wmma_f32_16x16x4_f32(
                false, a, false, bf, (short)0, acc, false, false);

            int cn = sj * 16 + nn;
            float bx1 = cB[cn * 4 + 0], by1 = cB[cn * 4 + 1];
            float bx2 = cB[cn * 4 + 2], by2 = cB[cn * 4 + 3];
            #pragma unroll
            for (int r = 0; r < 8; ++r) {
                int m = si * 16 + r + half * 8;   // D: M=r (+8 for hi half-wave)
                float ax1 = rB[m * 4 + 0], ay1 = rB[m * 4 + 1];
                float ax2 = rB[m * 4 + 2], ay2 = rB[m * 4 + 3];
                float ix = fminf(ax2, bx2) - fmaxf(ax1, bx1);
                float iy = fminf(ay2, by2) - fmaxf(ay1, by1);
                float inter = fmaxf(ix, 0.f) * fmaxf(iy, 0.f);
                int gi = ti * 32 + m, gj = tj * 32 + cn;
                // iou > thr  <=>  inter > thr*(union), union = acc - inter (>0)
                bool bit = (inter > IOU_THR * (acc[r] - inter)) && (gj > gi);
                unsigned bal = (unsigned)__ballot(bit); // lo16=row, hi16=row+8
                int rowLo = ti * 32 + si * 16 + r;
                int wordJ = tj * 2 + sj;
                if (lane == 0)
                    mask[((size_t)b * MROWS + rowLo) * 128 + wordJ] = bal & 0xFFFFu;
                if (lane == 16)
                    mask[((size_t)b * MROWS + rowLo + 8) * 128 + wordJ] = (bal >> 16) & 0xFFFFu;
            }
        }
    }
}

// -------- kernel 5: sequential NMS over LDS bitmap + top-100 ---------
__global__ __launch_bounds__(128)
void nms_kernel(const float* __restrict__ selScore,
                const float* __restrict__ selBox,
                const int*   __restrict__ selLabel,
                const unsigned* __restrict__ mask,
                float* __restrict__ out) {
    const int b = blockIdx.x;
    const int tid = threadIdx.x;                 // 128 threads = 128 mask words
    __shared__ unsigned keepW[128];
    __shared__ int detIdx[MAXDET];
    __shared__ int detCnt;

    unsigned wbits = 0;
    for (int k = 0; k < 16; ++k) {
        int j = tid * 16 + k;
        if (j < KPRE && selScore[b * KSORT + j] > SCORE_THR) wbits |= (1u << k);
    }
    keepW[tid] = wbits;
    __syncthreads();

    for (int i = 0; i < KPRE; ++i) {
        unsigned kept = (keepW[i >> 4] >> (i & 15)) & 1u;
        // mask row i never contains bits j<=i, so the concurrent update below
        // cannot change the bit read above; one barrier per iteration suffices.
        if (kept && tid >= (i >> 4))
            keepW[tid] &= ~mask[((size_t)b * MROWS + i) * 128 + tid];
        __syncthreads();
    }

    if (tid == 0) {        // kept list is already score-descending
        int cnt = 0;
        for (int j = 0; j < KPRE && cnt < MAXDET; ++j)
            if ((keepW[j >> 4] >> (j & 15)) & 1u) detIdx[cnt++] = j;
        detCnt = cnt;
    }
    __syncthreads();

    float* oBox   = out;                               // [B,100,4]
    float* oScore = out + BATCH * MAXDET * 4;          // [B,100]
    int*   oLabel = (int*)(out + BATCH * MAXDET * 5);  // [B,100] int32 bits

    if (tid < MAXDET) {
        int o = b * MAXDET + tid;
        if (tid < detCnt) {
            int j = detIdx[tid];
            const float* p = selBox + ((size_t)b * KSORT + j) * 4;
            oBox[o * 4 + 0] = p[0]; oBox[o * 4 + 1] = p[1];
            oBox[o * 4 + 2] = p[2]; oBox[o * 4 + 3] = p[3];
            oScore[o] = selScore[b * KSORT + j];
            oLabel[o] = selLabel[b * KSORT + j];
        } else {
            oBox[o * 4 + 0] = 0.f; oBox[o * 4 + 1] = 0.f;
            oBox[o * 4 + 2] = 0.f; oBox[o * 4 + 3] = 0.f;
            oScore[o] = 0.f;
            oLabel[o] = -1;
        }
    }
}

// ---------------------------------------------------------------------
extern "C" void kernel_launch(void* const* d_in, const int* in_sizes, int n_in,
                              void* d_out, int out_size, void* d_ws, size_t ws_size,
                              hipStream_t stream) {
    const float* cls   = (const float*)d_in[0];  // [B,N,81]
    const float* reg   = (const float*)d_in[1];  // [B,N,320]
    const float* props = (const float*)d_in[2];  // [B,N,4]
    const int*   hw    = (const int*)d_in[3];    // [B,2]
    float* out = (float*)d_out;

    char* ws = (char*)d_ws;
    size_t off = 0;
    auto carve = [&](size_t bytes) -> char* {
        char* p = ws + off;
        off = (off + bytes + 255) & ~(size_t)255;
        return p;
    };
    float* cand    = (float*)carve((size_t)BATCH * NPROP * NCLS * 4);
    float* fbox    = (float*)carve((size_t)BATCH * NPROP * NCLS * 4 * 4);
    float* rowMax  = (float*)carve((size_t)BATCH * NPROP * 4);
    float* rowSum  = (float*)carve((size_t)BATCH * NPROP * 4);
    float* selScore= (float*)carve((size_t)BATCH * KSORT * 4);
    float* selBox  = (float*)carve((size_t)BATCH * KSORT * 4 * 4);
    int*   selLabel= (int*)  carve((size_t)BATCH * KSORT * 4);
    unsigned* mask = (unsigned*)carve((size_t)BATCH * MROWS * 128 * 4);

    rowstat_kernel<<<(BATCH * NPROP + 255) / 256, 256, 0, stream>>>(cls, rowMax, rowSum);
    score_decode_kernel<<<(BATCH * NPROP * NCLS + 255) / 256, 256, 0, stream>>>(
        cls, reg, props, hw, rowMax, rowSum, cand, fbox);
    select_kernel<<<BATCH, 1024, 0, stream>>>(cand, fbox, selScore, selBox, selLabel);
    dim3 gTiles(NT32, NT32, BATCH);
    iou_mask_kernel<<<gTiles, 32, 0, stream>>>(selBox, selLabel, hw, mask);
    nms_kernel<<<BATCH, 128, 0, stream>>>(selScore, selBox, selLabel, mask, out);
}